// HOGGeneratorMel_56994216018056
// MI455X (gfx1250) — compile-verified
//
#include <hip/hip_runtime.h>
#include <stdint.h>

#define NBINS   108
#define POOL_F  16
#define POOL_T  25
#define FDIM    128
#define TDIM    2000
#define NF      8          // 128 / 16
#define NT      80         // 2000 / 25
#define TILE_R  18         // 16 + 2 halo
#define TILE_C  27         // 25 + 2 halo
#define TILE_S  28         // padded LDS row stride
#define BLOCK   128        // 4 wave32s

typedef __attribute__((address_space(1))) int g_int;   // global int
typedef __attribute__((address_space(3))) int l_int;   // LDS int

__global__ __launch_bounds__(BLOCK)
void hog_patch_kernel(const float* __restrict__ x, float* __restrict__ out) {
    __shared__ float tile[TILE_R * TILE_S];
    __shared__ float hist[NBINS];
    __shared__ float red[BLOCK];

    const int tid = threadIdx.x;
    const int blk = blockIdx.x;
    const int ti  = blk % NT;
    const int fi  = (blk / NT) % NF;
    const int b   = blk / (NT * NF);
    const int r0  = fi * POOL_F;
    const int c0  = ti * POOL_T;

    if (tid < NBINS) hist[tid] = 0.0f;

    const float* xb = x + (size_t)b * FDIM * TDIM;

    // ---- Stage 18x27 halo tile into LDS (reflect padding at image edges) ----
    for (int i = tid; i < TILE_R * TILE_C; i += BLOCK) {
        const int tr = i / TILE_C;
        const int tc = i - tr * TILE_C;
        int gr = r0 - 1 + tr;
        int gc = c0 - 1 + tc;
        gr = (gr < 0) ? 1 : ((gr >= FDIM) ? (2 * FDIM - 2 - gr) : gr);
        gc = (gc < 0) ? 1 : ((gc >= TDIM) ? (2 * TDIM - 2 - gc) : gc);
        const float* gp = xb + (size_t)gr * TDIM + gc;
        float*       lp = &tile[tr * TILE_S + tc];
#if defined(__gfx1250__) && __has_builtin(__builtin_amdgcn_global_load_async_to_lds_b32)
        // CDNA5 async global->LDS gather (per-lane), tracked by ASYNCcnt.
        __builtin_amdgcn_global_load_async_to_lds_b32(
            (g_int*)(void*)gp, (l_int*)(void*)lp, 0, 0);
#else
        *lp = *gp;
#endif
    }
#if defined(__gfx1250__) && __has_builtin(__builtin_amdgcn_global_load_async_to_lds_b32)
#if __has_builtin(__builtin_amdgcn_s_wait_asynccnt)
    __builtin_amdgcn_s_wait_asynccnt(0);
#else
    asm volatile("s_wait_asynccnt 0" ::: "memory");
#endif
#endif
    __syncthreads();

    // ---- Sobel + orientation binning + LDS histogram (ds_add_f32) ----
    const float PI_F = 3.14159265358979323846f;
    for (int e = tid; e < POOL_F * POOL_T; e += BLOCK) {
        const int row = e / POOL_T;
        const int col = e - row * POOL_T;
        const float* t0 = &tile[row * TILE_S + col];
        const float a00 = t0[0],           a01 = t0[1],            a02 = t0[2];
        const float a10 = t0[TILE_S],                              a12 = t0[TILE_S + 2];
        const float a20 = t0[2 * TILE_S],  a21 = t0[2 * TILE_S+1], a22 = t0[2 * TILE_S + 2];
        // cross-correlation with sobel / sobel^T (matches lax.conv semantics)
        const float gx = (a00 - a02) + 2.0f * (a10 - a12) + (a20 - a22);
        const float gy = (a00 - a20) + 2.0f * (a01 - a21) + (a02 - a22);
        const float mag = sqrtf(gx * gx + gy * gy);
        float ph = atan2f(gx, gy) / PI_F;   // arctan2(gx, gy) / pi
        ph = ph * (float)NBINS;
        int p = (int)floorf(ph);
        int bin = p % NBINS;
        if (bin < 0) bin += NBINS;          // Python-style mod (sign of divisor)
        atomicAdd(&hist[bin], mag);         // ds_add_f32 (no return)
    }
    __syncthreads();

    // ---- L2 norm over the 108 bins, then normalized write-out ----
    float v = 0.0f;
    if (tid < NBINS) { const float h = hist[tid]; v = h * h; }
    red[tid] = v;
    __syncthreads();
    for (int s = BLOCK / 2; s > 0; s >>= 1) {
        if (tid < s) red[tid] += red[tid + s];
        __syncthreads();
    }
    const float inv = 1.0f / fmaxf(sqrtf(red[0]), 1e-12f);
    if (tid < NBINS) out[(size_t)blk * NBINS + tid] = hist[tid] * inv;
}

extern "C" void kernel_launch(void* const* d_in, const int* in_sizes, int n_in,
                              void* d_out, int out_size, void* d_ws, size_t ws_size,
                              hipStream_t stream) {
    (void)n_in; (void)out_size; (void)d_ws; (void)ws_size;
    const float* x  = (const float*)d_in[0];
    float*       out = (float*)d_out;
    const int B = in_sizes[0] / (FDIM * TDIM);   // 8 for the reference shapes
    const int nblocks = B * NF * NT;             // one block per 16x25 patch
    hog_patch_kernel<<<dim3(nblocks), dim3(BLOCK), 0, stream>>>(x, out);
}